// MultiOmicsGenerator_33071248179786
// MI455X (gfx1250) — compile-verified
//
#include <hip/hip_runtime.h>
#include <math.h>

typedef float v2f __attribute__((ext_vector_type(2)));
typedef float v4f __attribute__((ext_vector_type(4)));
typedef float v8f __attribute__((ext_vector_type(8)));

#define LATD 512
#define BNEPS 1e-3f

// ---------------------------------------------------------------------------
// Step 0: row-mean of the [4, 512] latent matrix (GCN collapses to this).
// ---------------------------------------------------------------------------
__global__ void mean4_kernel(const float* __restrict__ L, float* __restrict__ xbar) {
    int j = blockIdx.x * blockDim.x + threadIdx.x;
    if (j < LATD)
        xbar[j] = 0.25f * (L[j] + L[LATD + j] + L[2 * LATD + j] + L[3 * LATD + j]);
}

// ---------------------------------------------------------------------------
// Wave-level fp32 GEMV via V_WMMA_F32_16X16X4_F32.
// Computes y[jbase + 0..15] = sum_k x[k] * W[k*512 + j].
// A-tile (16x4 f32): only row M=0 is live. Per ISA layout, lane 0 carries
// (K=0,K=1) in (v0,v1) and lane 16 carries (K=2,K=3); all other lanes zero.
// B-tile (4x16 f32): v0 holds rows K={0 | 2} and v1 rows K={1 | 3} across the
// two lane halves (same striping rule as the documented C/D layout).
// D row M=0 lives in c[0] on lanes 0..15.
// Must be called with blockDim.x == 32 (full wave, EXEC all ones).
// ---------------------------------------------------------------------------
__device__ __forceinline__ float gemv16_wmma(const float* __restrict__ x,
                                             const float* __restrict__ W,
                                             int jbase) {
    const int lane = threadIdx.x & 31;
    const int half = lane >> 4;     // 0: lanes 0-15, 1: lanes 16-31
    const int col  = lane & 15;
    const int j    = jbase + col;
    v8f c0 = {}; v8f c1 = {};       // two independent accumulation chains (ILP)
    for (int k0 = 0; k0 < LATD; k0 += 8) {
        {
            const int kb = k0 + 2 * half;
            v2f a, b;
            const float a0 = x[kb], a1 = x[kb + 1];
            a.x = (col == 0) ? a0 : 0.0f;     // only M=0 carrier lanes (0, 16)
            a.y = (col == 0) ? a1 : 0.0f;
            b.x = W[(size_t)kb * LATD + j];
            b.y = W[(size_t)(kb + 1) * LATD + j];
            c0 = __builtin_amdgcn_wmma_f32_16x16x4_f32(false, a, false, b,
                                                       (short)0, c0, false, false);
        }
        {
            const int kb = k0 + 4 + 2 * half;
            v2f a, b;
            const float a0 = x[kb], a1 = x[kb + 1];
            a.x = (col == 0) ? a0 : 0.0f;
            a.y = (col == 0) ? a1 : 0.0f;
            b.x = W[(size_t)kb * LATD + j];
            b.y = W[(size_t)(kb + 1) * LATD + j];
            c1 = __builtin_amdgcn_wmma_f32_16x16x4_f32(false, a, false, b,
                                                       (short)0, c1, false, false);
        }
    }
    return c0[0] + c1[0];
}

// GCN layer (collapsed): y = relu(x @ W).  grid=32 blocks x 1 wave.
__global__ void gcn_gemv_kernel(const float* __restrict__ x,
                                const float* __restrict__ W,
                                float* __restrict__ y) {
    const int jbase = blockIdx.x * 16;
    const float r = gemv16_wmma(x, W, jbase);
    const int lane = threadIdx.x & 31;
    if (lane < 16) y[jbase + lane] = fmaxf(r, 0.0f);
}

// Generator stage 1: h[t] = relu(BN(x @ W1[t] + b1[t])).  grid=(32,4).
__global__ void dense1_kernel(const float* __restrict__ x,
                              const float* __restrict__ W1,
                              const float* __restrict__ b1,
                              const float* __restrict__ g1,
                              const float* __restrict__ be1,
                              const float* __restrict__ m1,
                              const float* __restrict__ v1,
                              float* __restrict__ h) {
    const int t = blockIdx.y;
    const int jbase = blockIdx.x * 16;
    const float acc = gemv16_wmma(x, W1 + (size_t)t * LATD * LATD, jbase);
    const int lane = threadIdx.x & 31;
    if (lane < 16) {
        const int j  = jbase + lane;
        const int pj = t * LATD + j;
        const float yv = acc + b1[pj];
        const float bn = g1[pj] * (yv - m1[pj]) * rsqrtf(v1[pj] + BNEPS) + be1[pj];
        h[(size_t)t * LATD + j] = fmaxf(bn, 0.0f);
    }
}

// ---------------------------------------------------------------------------
// Heads: out = BN(h_node @ W2 + b2).  Bandwidth-bound (506 MB of W2 total):
// each thread owns 4 consecutive columns -> coalesced b128 streaming loads
// with NT hint (W2 >> L2).  h vector broadcast from LDS.
// ---------------------------------------------------------------------------
__global__ __launch_bounds__(256) void gen_out_kernel(
        const float* __restrict__ hvec, const float* __restrict__ W2,
        const float* __restrict__ b2,  const float* __restrict__ g2,
        const float* __restrict__ be2, const float* __restrict__ m2,
        const float* __restrict__ v2,  float* __restrict__ out, int D) {
    __shared__ float hs[LATD];
    for (int i = threadIdx.x; i < LATD; i += 256) hs[i] = hvec[i];
    __syncthreads();

    const int c4 = (blockIdx.x * 256 + threadIdx.x) * 4;   // D % 4 == 0 always
    if (c4 >= D) return;

    const v4f* Wp = (const v4f*)(W2 + c4);
    const size_t strideV4 = (size_t)D >> 2;
    v4f acc = {};
#pragma unroll 8
    for (int k = 0; k < LATD; ++k) {
        const v4f w = __builtin_nontemporal_load(Wp + (size_t)k * strideV4);
        const float hk = hs[k];
        acc.x = fmaf(hk, w.x, acc.x);
        acc.y = fmaf(hk, w.y, acc.y);
        acc.z = fmaf(hk, w.z, acc.z);
        acc.w = fmaf(hk, w.w, acc.w);
    }
    const v4f bb = *(const v4f*)(b2  + c4);
    const v4f gg = *(const v4f*)(g2  + c4);
    const v4f ee = *(const v4f*)(be2 + c4);
    const v4f mm = *(const v4f*)(m2  + c4);
    const v4f vv = *(const v4f*)(v2  + c4);
    v4f y;
    y.x = gg.x * (acc.x + bb.x - mm.x) * rsqrtf(vv.x + BNEPS) + ee.x;
    y.y = gg.y * (acc.y + bb.y - mm.y) * rsqrtf(vv.y + BNEPS) + ee.y;
    y.z = gg.z * (acc.z + bb.z - mm.z) * rsqrtf(vv.z + BNEPS) + ee.z;
    y.w = gg.w * (acc.w + bb.w - mm.w) * rsqrtf(vv.w + BNEPS) + ee.w;
    *(v4f*)(out + c4) = y;
}

// ---------------------------------------------------------------------------
extern "C" void kernel_launch(void* const* d_in, const int* in_sizes, int n_in,
                              void* d_out, int out_size, void* d_ws, size_t ws_size,
                              hipStream_t stream) {
    const float* latent = (const float*)d_in[0];
    const float* Wg1    = (const float*)d_in[1];
    const float* Wg2    = (const float*)d_in[2];
    const float* W1     = (const float*)d_in[3];
    const float* b1     = (const float*)d_in[4];
    const float* g1     = (const float*)d_in[5];
    const float* be1    = (const float*)d_in[6];
    const float* m1     = (const float*)d_in[7];
    const float* v1     = (const float*)d_in[8];

    float* out  = (float*)d_out;
    float* ws   = (float*)d_ws;
    float* xbar = ws;              // 512
    float* z1   = ws + 512;        // 512
    float* z2   = ws + 1024;       // 512
    float* h    = ws + 1536;       // 4 x 512

    mean4_kernel<<<1, 512, 0, stream>>>(latent, xbar);
    gcn_gemv_kernel<<<32, 32, 0, stream>>>(xbar, Wg1, z1);
    gcn_gemv_kernel<<<32, 32, 0, stream>>>(z1, Wg2, z2);
    dense1_kernel<<<dim3(32, 4), 32, 0, stream>>>(z2, W1, b1, g1, be1, m1, v1, h);

    const int dims[4] = {20000, 200000, 2000, 25000};
    int off = 0;
    for (int o = 0; o < 4; ++o) {
        const float* W2  = (const float*)d_in[9 + 6 * o + 0];
        const float* b2  = (const float*)d_in[9 + 6 * o + 1];
        const float* g2  = (const float*)d_in[9 + 6 * o + 2];
        const float* be2 = (const float*)d_in[9 + 6 * o + 3];
        const float* m2  = (const float*)d_in[9 + 6 * o + 4];
        const float* v2  = (const float*)d_in[9 + 6 * o + 5];
        const int D = dims[o];
        const int blocks = (D / 4 + 255) / 256;
        gen_out_kernel<<<blocks, 256, 0, stream>>>(h + (size_t)o * LATD,
                                                   W2, b2, g2, be2, m2, v2,
                                                   out + off, D);
        off += D;
    }
}